// kernel_generated_2_21887153341291
// MI455X (gfx1250) — compile-verified
//
#include <hip/hip_runtime.h>

typedef __attribute__((ext_vector_type(2))) float v2f;
typedef __attribute__((ext_vector_type(8))) float v8f;

#define IC   128
#define OC   32
#define H    14
#define W    14
#define NSP  196            // H*W spatial positions
#define KTOT 384            // 3*IC reduction length
#define NT   13             // ceil(196/16) N tiles
#define MT   2              // 32/16 M tiles
#define WAVES (NT * MT)     // 26 waves, one 16x16 tile each
#define THREADS (WAVES * 32)

#define SX_COLS 16          // halo'd column index c <-> ww = c-1; c=0,15 are zero pad
#define SX_ROWS (H * SX_COLS)   // 224 (h,c) rows
#define SX_RSTR 130         // dwords per (h,c) row (128 ch + 2 pad; bank step 2)

// Single workgroup on one WGP. Rolled + zero-padded input staged K(=ic)-fastest
// in LDS so each WMMA B fragment is ONE aligned ds_load_b64 into a contiguous
// VGPR pair with an immediate offset (no per-step address math, no reg moves).
// Weights come in as cached global_load_b64 fragments. 26 waves x one 16x16
// tile x 96 accumulating exact-f32 WMMAs; EXEC stays all-ones in the hot loop.
__global__ __launch_bounds__(THREADS, 1)
void conv_roll_wmma(const float* __restrict__ x, const float* __restrict__ w,
                    float* __restrict__ out) {
  __shared__ float sxp[SX_ROWS * SX_RSTR];  // [h][c][ic], 116,480 B

  const int tid = threadIdx.x;

  // ---- Stage input: both circular rolls + zero halo, transposed to [h,c][ic]
  // sxp[(h*16+c)*130 + ic] = (1 <= c <= 14) ? x[ic][(h-1)%14][(c-2)%14] : 0
  // (ic-major enumeration keeps the GLOBAL reads coalesced; LDS writes scatter)
  for (int idx = tid; idx < IC * SX_ROWS; idx += THREADS) {
    const int ic  = idx / SX_ROWS;
    const int rem = idx % SX_ROWS;
    const int h   = rem / SX_COLS;
    const int c   = rem % SX_COLS;
    const int hs  = (h + H - 1) % H;
    const int ws  = (c + W - 2) % W;           // (ww-1) mod 14, ww = c-1
    const bool inb = (c >= 1) && (c <= W);
    sxp[rem * SX_RSTR + ic] = inb ? x[ic * NSP + hs * W + ws] : 0.0f;
  }
  __syncthreads();

  const int wave     = tid >> 5;
  const int lane     = tid & 31;
  const int laneHalf = lane & 15;
  const int gK       = (lane >> 4) << 1;  // K sub-offset of this half-wave: 0 or 2

  const int m0 = (wave & 1) * 16;         // output-channel tile base
  const int n0 = (wave >> 1) * 16;        // spatial tile base

  const int  n      = n0 + laneHalf;      // this lane's output column (N)
  const bool nvalid = (n < NSP);
  const int  nc     = nvalid ? n : (NSP - 1);  // clamp: junk columns, never stored
  const int  h      = nc / W;
  const int  wp     = nc % W;
  const int  coBase = h * SX_COLS + wp;   // (h,c) row index; +k selects the tap

  // A rows for this lane: w[m0+laneHalf][*]; kg even -> 8B-aligned b64 pairs.
  const v2f* __restrict__ aRow =
      reinterpret_cast<const v2f*>(w + (m0 + laneHalf) * KTOT);

  v8f c = {};
#pragma unroll
  for (int k = 0; k < 3; ++k) {
    // Per-k base: per-lane row of 128 channels; inner offsets are imm-sized.
    const float* __restrict__ bRow = &sxp[(coBase + k) * SX_RSTR + gK];
    for (int icb = 0; icb < IC; icb += 4) {
      const int kk0 = k * IC + icb;
      // A fragment: lane -> M, (VGPR, half-wave) -> K; one cached global b64.
      const v2f a = aRow[(kk0 + gK) >> 1];
      // B fragment: K-pair adjacent in LDS -> one aligned ds_load_b64.
      const v2f b = *reinterpret_cast<const v2f*>(bRow + icb);
      // D = A*B + C, exact f32 -> v_wmma_f32_16x16x4_f32
      c = __builtin_amdgcn_wmma_f32_16x16x4_f32(false, a, false, b,
                                                (short)0, c, false, false);
    }
  }

  // C/D layout: VGPR r -> M = m0 + r (+8 for lanes 16..31), lane -> N.
  if (nvalid) {
    const int ocBase = m0 + ((lane >> 4) << 3);
#pragma unroll
    for (int r = 0; r < 8; ++r) {
      out[(ocBase + r) * NSP + n] = c[r];
    }
  }
}

extern "C" void kernel_launch(void* const* d_in, const int* in_sizes, int n_in,
                              void* d_out, int out_size, void* d_ws, size_t ws_size,
                              hipStream_t stream) {
  (void)in_sizes; (void)n_in; (void)out_size; (void)d_ws; (void)ws_size;
  const float* x = (const float*)d_in[0];   // (1,128,14,14) f32
  const float* w = (const float*)d_in[1];   // (32,3,128)    f32
  float* out = (float*)d_out;               // (1,32,14,14)  f32
  conv_roll_wmma<<<dim3(1), dim3(THREADS), 0, stream>>>(x, w, out);
}